// GATPPI_16097537425783
// MI455X (gfx1250) — compile-verified
//
#include <hip/hip_runtime.h>
#include <hip/hip_bf16.h>
#include <math.h>

typedef __attribute__((ext_vector_type(2))) float v2f;
typedef __attribute__((ext_vector_type(8))) float v8f;
typedef int v4i __attribute__((vector_size(16)));
typedef __attribute__((address_space(1))) v4i* as1_v4i;
typedef __attribute__((address_space(3))) v4i* as3_v4i;

#define NN   100000
#define EE   1600000
#define BB   64
#define LL   1200
#define DD   128
#define MT_TAG (NN / 16)   // 6250, exact
#define SLAB 64            // K-rows of B staged in LDS per iteration (64*128*4 = 32KB)

#if defined(__has_builtin)
#if __has_builtin(__builtin_amdgcn_global_load_async_to_lds_b128) && \
    __has_builtin(__builtin_amdgcn_s_wait_asynccnt)
#define USE_ASYNC_LDS 1
#endif
#endif

__device__ inline v8f wmma_f32_4(v2f a, v2f b, v8f c) {
    // V_WMMA_F32_16X16X4_F32 : D = A(16x4) * B(4x16) + C(16x16)
    return __builtin_amdgcn_wmma_f32_16x16x4_f32(
        false, a, false, b, (short)0, c, false, false);
}

// Cooperatively stage SLAB x 128 floats of B (already offset to slab start) into LDS.
// 256 threads, each moves 32 floats as 8x b128 transfers.
__device__ inline void stage_slab(const float* __restrict__ Wslab, float* Bs, int tid) {
#pragma unroll
    for (int base = 0; base < SLAB * DD; base += 256 * 4) {
        int idx = base + tid * 4;
#ifdef USE_ASYNC_LDS
        __builtin_amdgcn_global_load_async_to_lds_b128(
            (as1_v4i)(Wslab + idx), (as3_v4i)(Bs + idx), 0, 0);
#else
        *(float4*)(Bs + idx) = *(const float4*)(Wslab + idx);
#endif
    }
}

__device__ inline void stage_wait() {
#ifdef USE_ASYNC_LDS
    __builtin_amdgcn_s_wait_asynccnt(0);
#endif
}

// ---------------- degree / norm ----------------
__global__ void k_deg(const int* __restrict__ dst, unsigned* __restrict__ deg, int E) {
    int e = blockIdx.x * blockDim.x + threadIdx.x;
    if (e < E) atomicAdd(&deg[dst[e]], 1u);
}

__global__ void k_norm(float* __restrict__ buf, int n) {
    int i = blockIdx.x * blockDim.x + threadIdx.x;
    if (i < n) {
        unsigned d = ((const unsigned*)buf)[i];
        float dv = fmaxf((float)d, 1.0f);
        buf[i] = rsqrtf(dv);
    }
}

// ---------------- one TAGConv hop: atomicAdd( next[dst], norm[src]*prev[src] ) ----------------
__global__ void k_hop(const float* __restrict__ prev, const int* __restrict__ src,
                      const int* __restrict__ dst, const float* __restrict__ norm,
                      float* __restrict__ next, int E) {
    int e = blockIdx.x * (blockDim.x >> 5) + (threadIdx.x >> 5);
    if (e >= E) return;
    int lane = threadIdx.x & 31;
    int s = src[e], d = dst[e];
    float ns = norm[s];
    const float4 p = *(const float4*)&prev[(size_t)s * DD + lane * 4];
    float* q = &next[(size_t)d * DD + lane * 4];
    atomicAdd(q + 0, ns * p.x);
    atomicAdd(q + 1, ns * p.y);
    atomicAdd(q + 2, ns * p.z);
    atomicAdd(q + 3, ns * p.w);
}

__global__ void k_scale(float* __restrict__ cur, const float* __restrict__ norm, int total) {
    int i = blockIdx.x * blockDim.x + threadIdx.x;
    if (i < total) cur[i] *= norm[i >> 7];
}

// ---------------- TAG GEMM: relu([feat|cur1|cur2] @ W + b), fused segment_max -> gmax ----
// 256 threads = 8 waves; each wave one 16x128 tile; B staged per 64-row slab in LDS.
__global__ void k_tag_gemm(const float* __restrict__ feat, const float* __restrict__ cur1,
                           const float* __restrict__ cur2, const float* __restrict__ W,
                           const float* __restrict__ bias, const int* __restrict__ gids,
                           float* __restrict__ gmax) {
    __shared__ float Bs[SLAB * DD];
    int tid = threadIdx.x;
    int mtile = blockIdx.x * 8 + (tid >> 5);
    if (mtile >= MT_TAG) mtile = MT_TAG - 1;   // keep wave alive for barriers; dup work is idempotent
    int lane = tid & 31;
    int half = lane >> 4, lr = lane & 15;
    size_t arow = (size_t)(mtile * 16 + lr) * DD;

    v8f acc[8];
#pragma unroll
    for (int nt = 0; nt < 8; ++nt) acc[nt] = 0.0f;

    for (int ks = 0; ks < 384; ks += SLAB) {
        __syncthreads();
        stage_slab(W + (size_t)ks * DD, Bs, tid);
        stage_wait();
        __syncthreads();

        const float* srcp; int kbase;
        if (ks < 128)      { srcp = feat; kbase = ks; }
        else if (ks < 256) { srcp = cur1; kbase = ks - 128; }
        else               { srcp = cur2; kbase = ks - 256; }

#pragma unroll
        for (int kk = 0; kk < SLAB; kk += 4) {
            v2f a = *(const v2f*)&srcp[arow + kbase + kk + half * 2];
            int kb = (kk + half * 2) * DD;
#pragma unroll
            for (int nt = 0; nt < 8; ++nt) {
                int col = nt * 16 + lr;
                v2f b; b.x = Bs[kb + col]; b.y = Bs[kb + DD + col];
                acc[nt] = wmma_f32_4(a, b, acc[nt]);
            }
        }
    }
#pragma unroll
    for (int r = 0; r < 8; ++r) {
        int m = mtile * 16 + r + half * 8;
        int gid = gids[m];
#pragma unroll
        for (int nt = 0; nt < 8; ++nt) {
            int col = nt * 16 + lr;
            float v = fmaxf(acc[nt][r] + bias[col], 0.0f);
            atomicMax((unsigned*)&gmax[gid * DD + col], __float_as_uint(v));
        }
    }
}

// ---------------- conv1d-as-GEMM: out[b][l][o] = sum_k A[l][k]*Bm[k][o] + bias ----------
// 256 threads = 8 waves; wave w handles l-tile (blockIdx.x*8 + w); B staged in LDS.
__global__ void k_conv_gemm(const float* __restrict__ in, const float* __restrict__ Bm,
                            const float* __restrict__ bias, float* __restrict__ out,
                            int Lin, int Lout, int Mtiles) {
    __shared__ float Bs[SLAB * DD];
    int tid = threadIdx.x;
    int b = blockIdx.y;
    int mtile = blockIdx.x * 8 + (tid >> 5);
    if (mtile >= Mtiles) mtile = Mtiles - 1;   // duplicate tiles store identical values
    int l0 = mtile * 16;
    int lane = tid & 31;
    int half = lane >> 4, lr = lane & 15;
    const float* inb = in + (size_t)b * Lin * DD;

    v8f acc[8];
#pragma unroll
    for (int nt = 0; nt < 8; ++nt) acc[nt] = 0.0f;

    for (int ks = 0; ks < 384; ks += SLAB) {
        __syncthreads();
        stage_slab(Bm + (size_t)ks * DD, Bs, tid);
        stage_wait();
        __syncthreads();

#pragma unroll
        for (int kk = 0; kk < SLAB; kk += 4) {
            int k0 = ks + kk;
            int trow = k0 >> 7;
            int ch = (k0 & 127) + half * 2;
            int row = l0 + lr + trow;
            if (row > Lin - 1) row = Lin - 1;
            v2f a = *(const v2f*)&inb[(size_t)row * DD + ch];
            int kb = (kk + half * 2) * DD;
#pragma unroll
            for (int nt = 0; nt < 8; ++nt) {
                int col = nt * 16 + lr;
                v2f bf; bf.x = Bs[kb + col]; bf.y = Bs[kb + DD + col];
                acc[nt] = wmma_f32_4(a, bf, acc[nt]);
            }
        }
    }
#pragma unroll
    for (int r = 0; r < 8; ++r) {
        int rowo = l0 + r + half * 8;
        if (rowo >= Lout) continue;
        float* ob = out + ((size_t)b * Lout + rowo) * DD;
#pragma unroll
        for (int nt = 0; nt < 8; ++nt) {
            int col = nt * 16 + lr;
            ob[col] = acc[nt][r] + bias[col];
        }
    }
}

// ---------------- conv weight reorder: W[o,i,t] -> Bm[t*128+i][o] ----------------
__global__ void k_reorder(const float* __restrict__ W, float* __restrict__ Bm) {
    int idx = blockIdx.x * blockDim.x + threadIdx.x; // o*384 + i*3 + t
    if (idx >= 128 * 128 * 3) return;
    int o = idx / 384;
    int rem = idx - o * 384;
    int i = rem / 3;
    int t = rem - i * 3;
    Bm[(t * 128 + i) * 128 + o] = W[idx];
}

// ---------------- maxpool k=3 s=3 over length ----------------
__global__ void k_pool3(const float* __restrict__ in, float* __restrict__ out, int Lin, int Lp) {
    int idx = blockIdx.x * blockDim.x + threadIdx.x;
    int total = BB * Lp * DD;
    if (idx >= total) return;
    int c = idx & 127;
    int lp = (idx >> 7) % Lp;
    int b = idx / (Lp * DD);
    const float* p = in + ((size_t)b * Lin + 3 * lp) * DD + c;
    float v = fmaxf(p[0], fmaxf(p[DD], p[2 * DD]));
    out[idx] = v;
}

// ---------------- final pool (window 130 -> 1) ----------------
__global__ void k_finalpool(const float* __restrict__ in, float* __restrict__ out, int Lin) {
    int idx = blockIdx.x * blockDim.x + threadIdx.x;
    if (idx >= BB * DD) return;
    int c = idx & 127;
    int b = idx >> 7;
    const float* p = in + (size_t)b * Lin * DD + c;
    float v = -INFINITY;
    for (int l = 0; l < Lin; ++l) v = fmaxf(v, p[(size_t)l * DD]);
    out[idx] = v;
}

// ---------------- small dense: out[b,o] = (relu)(in[b,:] @ W + bias) ----------------
__global__ void k_fc(const float* __restrict__ in, const float* __restrict__ W,
                     const float* __restrict__ bias, float* __restrict__ out,
                     int rows, int Kin, int Kout, int relu) {
    int idx = blockIdx.x * blockDim.x + threadIdx.x;
    if (idx >= rows * Kout) return;
    int b = idx / Kout, o = idx - b * Kout;
    const float* ip = in + (size_t)b * Kin;
    float s = bias[o];
    for (int k = 0; k < Kin; ++k) s += ip[k] * W[(size_t)k * Kout + o];
    out[idx] = relu ? fmaxf(s, 0.0f) : s;
}

// ---------------- batchnorm (training stats over 64 rows) + relu, in place ----------------
__global__ void k_bn_relu(float* __restrict__ x, const float* __restrict__ g,
                          const float* __restrict__ bb, int C, int rows) {
    int col = blockIdx.x * blockDim.x + threadIdx.x;
    if (col >= C) return;
    float mu = 0.f;
    for (int r = 0; r < rows; ++r) mu += x[(size_t)r * C + col];
    mu /= (float)rows;
    float var = 0.f;
    for (int r = 0; r < rows; ++r) { float d = x[(size_t)r * C + col] - mu; var += d * d; }
    var /= (float)rows;
    float inv = rsqrtf(var + 1e-5f);
    float gg = g[col], bv = bb[col];
    for (int r = 0; r < rows; ++r) {
        float v = gg * (x[(size_t)r * C + col] - mu) * inv + bv;
        x[(size_t)r * C + col] = fmaxf(v, 0.0f);
    }
}

// ---------------- branch combine: gcat[b, off+c] = (1-w)*gfeat + w*seq ----------------
__global__ void k_combine(const float* __restrict__ gfeat, const float* __restrict__ seq,
                          const float* __restrict__ w1, float* __restrict__ gcat, int off) {
    int idx = blockIdx.x * blockDim.x + threadIdx.x;
    if (idx >= BB * DD) return;
    float w = 1.0f / (1.0f + __expf(-w1[0]));
    int b = idx >> 7, c = idx & 127;
    gcat[b * 256 + off + c] = (1.0f - w) * gfeat[idx] + w * seq[idx];
}

// ---------------- output head: sigmoid(y2 @ out_W + out_b) ----------------
__global__ void k_head(const float* __restrict__ y2, const float* __restrict__ W,
                       const float* __restrict__ bias, float* __restrict__ out) {
    int b = blockIdx.x * blockDim.x + threadIdx.x;
    if (b >= BB) return;
    const float* p = y2 + (size_t)b * 256;
    float s = bias[0];
    for (int k = 0; k < 256; ++k) s += p[k] * W[k];
    out[b] = 1.0f / (1.0f + __expf(-s));
}

extern "C" void kernel_launch(void* const* d_in, const int* in_sizes, int n_in,
                              void* d_out, int out_size, void* d_ws, size_t ws_size,
                              hipStream_t stream) {
    const float* feat[2]  = { (const float*)d_in[0], (const float*)d_in[5] };
    const float* pad[2]   = { (const float*)d_in[1], (const float*)d_in[6] };
    const int*   esrc[2]  = { (const int*)d_in[2],   (const int*)d_in[7] };
    const int*   edst[2]  = { (const int*)d_in[3],   (const int*)d_in[8] };
    const int*   gids[2]  = { (const int*)d_in[4],   (const int*)d_in[9] };
    const float* tagW  = (const float*)d_in[10];
    const float* tagB  = (const float*)d_in[11];
    const float* fcgW  = (const float*)d_in[12];
    const float* fcgB  = (const float*)d_in[13];
    const float* cW[3] = { (const float*)d_in[14], (const float*)d_in[16], (const float*)d_in[18] };
    const float* cB[3] = { (const float*)d_in[15], (const float*)d_in[17], (const float*)d_in[19] };
    const float* flatW = (const float*)d_in[20];
    const float* flatB = (const float*)d_in[21];
    const float* w1    = (const float*)d_in[22];
    const float* fc1W  = (const float*)d_in[23];
    const float* fc1B  = (const float*)d_in[24];
    const float* bn1g  = (const float*)d_in[25];
    const float* bn1b  = (const float*)d_in[26];
    const float* fc2W  = (const float*)d_in[27];
    const float* fc2B  = (const float*)d_in[28];
    const float* bn2g  = (const float*)d_in[29];
    const float* bn2b  = (const float*)d_in[30];
    const float* outW  = (const float*)d_in[31];
    const float* outB  = (const float*)d_in[32];

    // ----- workspace layout (float elements, 256-elem aligned) -----
    float* ws = (float*)d_ws;
    size_t off = 0;
    auto take = [&](size_t n) { size_t r = off; off += (n + 255) & ~(size_t)255; return r; };
    size_t o_norm  = take(NN);
    size_t o_cur1  = take((size_t)NN * DD);   // also conv scratch A
    size_t o_cur2  = take((size_t)NN * DD);   // also conv scratch B
    size_t o_gmax  = take(BB * DD);
    size_t o_gfeat = take(BB * DD);
    size_t o_seqv  = take(BB * DD);
    size_t o_seqb  = take(BB * DD);
    size_t o_gcat  = take(BB * 256);
    size_t o_y1    = take(BB * 512);
    size_t o_y2    = take(BB * 256);
    size_t o_bm[3] = { take(384 * 128), take(384 * 128), take(384 * 128) };

    float* normb = ws + o_norm;
    float* cur1  = ws + o_cur1;
    float* cur2  = ws + o_cur2;
    float* gmax  = ws + o_gmax;
    float* gfeat = ws + o_gfeat;
    float* seqv  = ws + o_seqv;
    float* seqb  = ws + o_seqb;
    float* gcat  = ws + o_gcat;
    float* y1    = ws + o_y1;
    float* y2    = ws + o_y2;

    // conv scratch aliases (out buffers in cur1 region, pooled buffers in cur2 region)
    float* convA = cur1;  // holds out1 (9.81M f) / out2 / out3
    float* convB = cur2;  // holds pool1 (3.27M f) / pool2

    // reorder conv weights once
    for (int j = 0; j < 3; ++j)
        k_reorder<<<(128 * 128 * 3 + 255) / 256, 256, 0, stream>>>(cW[j], ws + o_bm[j]);

    for (int br = 0; br < 2; ++br) {
        // zero accumulators
        (void)hipMemsetAsync(normb, 0, NN * sizeof(float), stream);
        (void)hipMemsetAsync(cur1, 0, (size_t)NN * DD * sizeof(float), stream);
        (void)hipMemsetAsync(cur2, 0, (size_t)NN * DD * sizeof(float), stream);
        (void)hipMemsetAsync(gmax, 0, BB * DD * sizeof(float), stream);

        // degrees -> norm
        k_deg<<<(EE + 255) / 256, 256, 0, stream>>>(edst[br], (unsigned*)normb, EE);
        k_norm<<<(NN + 255) / 256, 256, 0, stream>>>(normb, NN);

        // hop 1: feat -> cur1 ; hop 2: cur1 -> cur2
        k_hop<<<(EE + 7) / 8, 256, 0, stream>>>(feat[br], esrc[br], edst[br], normb, cur1, EE);
        k_scale<<<((size_t)NN * DD + 255) / 256, 256, 0, stream>>>(cur1, normb, NN * DD);
        k_hop<<<(EE + 7) / 8, 256, 0, stream>>>(cur1, esrc[br], edst[br], normb, cur2, EE);
        k_scale<<<((size_t)NN * DD + 255) / 256, 256, 0, stream>>>(cur2, normb, NN * DD);

        // TAG GEMM + relu + fused segment_max
        k_tag_gemm<<<(MT_TAG + 7) / 8, 256, 0, stream>>>(feat[br], cur1, cur2, tagW, tagB,
                                                         gids[br], gmax);
        // gfeat = relu(gmax @ fc_g1_W + b)
        k_fc<<<(BB * DD + 255) / 256, 256, 0, stream>>>(gmax, fcgW, fcgB, gfeat, BB, 128, 128, 1);

        // TextCNN chain (note: convA/convB reuse cur1/cur2 — cur* consumed above)
        float* out1 = convA;  // [B,1198,128]
        float* pl1  = convB;  // [B,399,128]
        float* out2 = convA;  // [B,397,128]
        float* pl2  = convB;  // [B,132,128]
        float* out3 = convA;  // [B,130,128]
        k_conv_gemm<<<dim3((75 + 7) / 8, BB), 256, 0, stream>>>(pad[br], ws + o_bm[0], cB[0],
                                                                out1, 1200, 1198, 75);
        k_pool3<<<((size_t)BB * 399 * DD + 255) / 256, 256, 0, stream>>>(out1, pl1, 1198, 399);
        k_conv_gemm<<<dim3((25 + 7) / 8, BB), 256, 0, stream>>>(pl1, ws + o_bm[1], cB[1],
                                                                out2, 399, 397, 25);
        k_pool3<<<((size_t)BB * 132 * DD + 255) / 256, 256, 0, stream>>>(out2, pl2, 397, 132);
        k_conv_gemm<<<dim3((9 + 7) / 8, BB), 256, 0, stream>>>(pl2, ws + o_bm[2], cB[2],
                                                               out3, 132, 130, 9);
        k_finalpool<<<(BB * DD + 255) / 256, 256, 0, stream>>>(out3, seqv, 130);
        k_fc<<<(BB * DD + 255) / 256, 256, 0, stream>>>(seqv, flatW, flatB, seqb, BB, 128, 128, 1);

        // combine into gcat[:, br*128 : br*128+128]
        k_combine<<<(BB * DD + 255) / 256, 256, 0, stream>>>(gfeat, seqb, w1, gcat, br * 128);
    }

    // head
    k_fc<<<(BB * 512 + 255) / 256, 256, 0, stream>>>(gcat, fc1W, fc1B, y1, BB, 256, 512, 0);
    k_bn_relu<<<2, 256, 0, stream>>>(y1, bn1g, bn1b, 512, BB);
    k_fc<<<(BB * 256 + 255) / 256, 256, 0, stream>>>(y1, fc2W, fc2B, y2, BB, 512, 256, 0);
    k_bn_relu<<<1, 256, 0, stream>>>(y2, bn2g, bn2b, 256, BB);
    k_head<<<2, 32, 0, stream>>>(y2, outW, outB, (float*)d_out);
}